// ElderGAT_82712480186383
// MI455X (gfx1250) — compile-verified
//
#include <hip/hip_runtime.h>
#include <hip/hip_bf16.h>

// ---------------------------------------------------------------------------
// ElderGAT on gfx1250 (MI455X).
// Dense projections use V_WMMA_F32_16X16X4_F32 (full-precision FP32 WMMA).
// Edge phase: 3 L2-resident passes (logits+segmax, exp+segsum, weighted scatter)
// using global_atomic_max_u32 (order-preserving float encoding) and
// global_atomic_add_f32. proj tables (<=38.4MB) sit in the 192MB L2.
// ---------------------------------------------------------------------------

typedef __attribute__((ext_vector_type(2))) float v2f;
typedef __attribute__((ext_vector_type(8))) float v8f;

__device__ __forceinline__ unsigned enc_f32(float x) {
  unsigned u = __float_as_uint(x);
  return (u >> 31) ? ~u : (u | 0x80000000u);
}
__device__ __forceinline__ float dec_f32(unsigned u) {
  unsigned b = (u >> 31) ? (u & 0x7fffffffu) : ~u;
  return __uint_as_float(b);
}

// ---------------------------------------------------------------------------
// C[M,Nout] = A[M,K] @ W[Nout,K]^T (+ bias[Nout]).  M,Nout multiples of 16,
// K multiple of 4.  One wave per 16x16 output tile, FP32 WMMA, K stepped by 4.
// A-matrix layout (ISA 7.12.2): lanes 0-15 hold K={kb,kb+1}, lanes 16-31 hold
// K={kb+2,kb+3}, row M = lane&15.  B symmetric with N = lane&15.
// C/D: VGPR v -> row v + 8*(lane>>4), col = lane&15.
// ---------------------------------------------------------------------------
__global__ void gat_wmma_gemm_f32(const float* __restrict__ A,
                                  const float* __restrict__ W,
                                  const float* __restrict__ bias,
                                  float* __restrict__ C,
                                  int mtiles, int ntiles, int K) {
  int wave = blockIdx.x * (blockDim.x >> 5) + (threadIdx.x >> 5);
  if (wave >= mtiles * ntiles) return;           // wave-uniform: EXEC stays full
  int lane = threadIdx.x & 31;
  int mt = wave / ntiles;
  int nt = wave - mt * ntiles;
  long long arow = (long long)(mt * 16 + (lane & 15)) * K;
  long long wrow = (long long)(nt * 16 + (lane & 15)) * K;
  int ksel = (lane >> 4) << 1;                   // 0 (lanes 0-15) or 2 (16-31)

  v8f acc = {0.f, 0.f, 0.f, 0.f, 0.f, 0.f, 0.f, 0.f};
  for (int kb = 0; kb < K; kb += 4) {
    v2f a, b;
    a.x = A[arow + kb + ksel];
    a.y = A[arow + kb + ksel + 1];
    b.x = W[wrow + kb + ksel];
    b.y = W[wrow + kb + ksel + 1];
    acc = __builtin_amdgcn_wmma_f32_16x16x4_f32(
        /*neg_a=*/false, a, /*neg_b=*/false, b,
        /*c_mod=*/(short)0, acc, /*reuse_a=*/false, /*reuse_b=*/false);
  }

  int col   = nt * 16 + (lane & 15);
  int rbase = mt * 16 + ((lane >> 4) << 3);
  long long Nout = (long long)ntiles * 16;
  float bv = bias ? bias[col] : 0.0f;
#pragma unroll
  for (int v = 0; v < 8; ++v)
    C[(long long)(rbase + v) * Nout + col] = acc[v] + bv;
}

// ---------------------------------------------------------------------------
__global__ void gat_fill_u32(unsigned* __restrict__ p, unsigned v, long long n) {
  long long i = (long long)blockIdx.x * blockDim.x + threadIdx.x;
  if (i < n) p[i] = v;
}

// Pass 1: per (edge, head) logit + segment-max into dst slots.
template <int HC>
__global__ void gat_edge_logits_max(const float* __restrict__ proj,
                                    const long long* __restrict__ ei,
                                    const float* __restrict__ eattr,
                                    const float* __restrict__ att,
                                    const float* __restrict__ eb,
                                    float* __restrict__ logits,
                                    unsigned* __restrict__ maxenc,
                                    long long E, long long N) {
  long long idx = (long long)blockIdx.x * blockDim.x + threadIdx.x;
  long long tot = (E + N) * 4;
  if (idx >= tot) return;
  long long e = idx >> 2;
  int h = (int)(idx & 3);
  long long s, d;
  float ea;
  if (e < E) {
    s = ei[e];
    d = ei[E + e];
    ea = eattr[2 * e] + 0.35f * eattr[2 * e + 1];
  } else {
    s = d = e - E;
    ea = 1.0f;
  }
  const float* ps = proj + s * (4 * HC) + h * HC;
  const float* pd = proj + d * (4 * HC) + h * HC;
  const float* at = att + h * HC;
  float sum = 0.0f;
#pragma unroll
  for (int c = 0; c < HC; ++c) {
    float v = ps[c] + pd[c];
    v = (v > 0.0f) ? v : 0.2f * v;              // leaky_relu(0.2)
    sum = fmaf(v, at[c], sum);
  }
  sum += ea * eb[h];
  logits[idx] = sum;
  atomicMax(&maxenc[d * 4 + h], enc_f32(sum));
}

// Pass 2: e = exp(logit - max[dst]); segment-sum into denom; store e in place.
__global__ void gat_edge_exp_sum(const long long* __restrict__ ei,
                                 float* __restrict__ logits,
                                 const unsigned* __restrict__ maxenc,
                                 float* __restrict__ denom,
                                 long long E, long long N) {
  long long idx = (long long)blockIdx.x * blockDim.x + threadIdx.x;
  long long tot = (E + N) * 4;
  if (idx >= tot) return;
  long long e = idx >> 2;
  int h = (int)(idx & 3);
  long long d = (e < E) ? ei[E + e] : (e - E);
  float m = dec_f32(maxenc[d * 4 + h]);
  float ex = __expf(logits[idx] - m);
  logits[idx] = ex;
  atomicAdd(&denom[d * 4 + h], ex);
}

// Pass 3: out[dst] += (e/denom[dst]) * proj[src], one thread per (edge, chan).
template <int HC>
__global__ void gat_edge_scatter(const float* __restrict__ proj,
                                 const long long* __restrict__ ei,
                                 const float* __restrict__ logits,
                                 const float* __restrict__ denom,
                                 float* __restrict__ out,
                                 long long E, long long N) {
  long long idx = (long long)blockIdx.x * blockDim.x + threadIdx.x;
  long long tot = (E + N) * (4 * HC);
  if (idx >= tot) return;
  long long e = idx / (4 * HC);
  int j = (int)(idx - e * (4 * HC));
  int h = j / HC;
  long long s, d;
  if (e < E) {
    s = ei[e];
    d = ei[E + e];
  } else {
    s = d = e - E;
  }
  float attn = logits[e * 4 + h] / fmaxf(denom[d * 4 + h], 1e-12f);
  atomicAdd(&out[d * (4 * HC) + j], attn * proj[s * (4 * HC) + j]);
}

__global__ void gat_elu_inplace(float* __restrict__ p, long long n) {
  long long i = (long long)blockIdx.x * blockDim.x + threadIdx.x;
  if (i >= n) return;
  float x = p[i];
  p[i] = (x > 0.0f) ? x : expm1f(x);
}

// Head-mean + residual(+bias, pre-added) + L2 normalize -> d_out[N,16].
__global__ void gat_finalize(const float* __restrict__ out2,
                             const float* __restrict__ res2out,
                             float* __restrict__ z, long long N) {
  long long n = (long long)blockIdx.x * blockDim.x + threadIdx.x;
  if (n >= N) return;
  float v[16];
  float ss = 0.0f;
#pragma unroll
  for (int c = 0; c < 16; ++c) {
    float a = out2[n * 64 + c] + out2[n * 64 + 16 + c] +
              out2[n * 64 + 32 + c] + out2[n * 64 + 48 + c];
    float val = 0.25f * a + res2out[n * 16 + c];
    v[c] = val;
    ss = fmaf(val, val, ss);
  }
  float inv = 1.0f / fmaxf(sqrtf(ss), 1e-12f);
#pragma unroll
  for (int c = 0; c < 16; ++c) z[n * 16 + c] = v[c] * inv;
}

// ---------------------------------------------------------------------------
extern "C" void kernel_launch(void* const* d_in, const int* in_sizes, int n_in,
                              void* d_out, int out_size, void* d_ws, size_t ws_size,
                              hipStream_t stream) {
  (void)n_in; (void)out_size; (void)ws_size;
  const float*     x     = (const float*)d_in[0];
  const long long* ei    = (const long long*)d_in[1];   // int64 [2,E]
  const float*     eattr = (const float*)d_in[2];       // [E,2]
  const float*     W1    = (const float*)d_in[3];
  const float*     att1  = (const float*)d_in[4];
  const float*     eb1   = (const float*)d_in[5];
  const float*     bias1 = (const float*)d_in[6];
  const float*     res1  = (const float*)d_in[7];
  const float*     W2    = (const float*)d_in[8];
  const float*     att2  = (const float*)d_in[9];
  const float*     eb2   = (const float*)d_in[10];
  const float*     bias2 = (const float*)d_in[11];
  const float*     res2  = (const float*)d_in[12];
  float* zout = (float*)d_out;

  const long long N = in_sizes[0] / 128;   // 50000
  const long long E = in_sizes[1] / 2;     // 800000
  const long long Ep = E + N;              // with self loops

  // --- workspace carve-out (256B aligned) ---------------------------------
  char* ws = (char*)d_ws;
  size_t off = 0;
  auto alloc = [&](size_t bytes) {
    char* p = ws + off;
    off = (off + bytes + 255) & ~(size_t)255;
    return p;
  };
  float*    proj1   = (float*)alloc((size_t)N * 192 * 4);   // 38.4 MB
  float*    out1    = (float*)alloc((size_t)N * 192 * 4);   // 38.4 MB (-> h)
  float*    logits  = (float*)alloc((size_t)Ep * 4 * 4);    // 13.6 MB (shared)
  unsigned* maxenc  = (unsigned*)alloc((size_t)N * 4 * 4);
  float*    denom   = (float*)alloc((size_t)N * 4 * 4);
  float*    proj2   = (float*)alloc((size_t)N * 64 * 4);    // 12.8 MB
  float*    out2    = (float*)alloc((size_t)N * 64 * 4);    // 12.8 MB
  float*    res2out = (float*)alloc((size_t)N * 16 * 4);    // 3.2 MB

  const int B = 256;
  const int wavesPerBlk = B / 32;
  auto gblk = [&](long long work) { return (unsigned)((work + B - 1) / B); };

  const int mtiles = (int)(N / 16);        // 3125 (N divisible by 16)

  // ===== layer 1 ===========================================================
  {
    int tiles = mtiles * 12;               // Nout = 192
    gat_wmma_gemm_f32<<<(tiles + wavesPerBlk - 1) / wavesPerBlk, B, 0, stream>>>(
        x, W1, nullptr, proj1, mtiles, 12, 128);
    gat_wmma_gemm_f32<<<(tiles + wavesPerBlk - 1) / wavesPerBlk, B, 0, stream>>>(
        x, res1, bias1, out1, mtiles, 12, 128);   // out1 = x@res1^T + bias1
  }
  gat_fill_u32<<<gblk(N * 4), B, 0, stream>>>(maxenc, 0u, N * 4);
  gat_fill_u32<<<gblk(N * 4), B, 0, stream>>>((unsigned*)denom, 0u, N * 4);

  gat_edge_logits_max<48><<<gblk(Ep * 4), B, 0, stream>>>(
      proj1, ei, eattr, att1, eb1, logits, maxenc, E, N);
  gat_edge_exp_sum<<<gblk(Ep * 4), B, 0, stream>>>(ei, logits, maxenc, denom, E, N);
  gat_edge_scatter<48><<<gblk(Ep * 192), B, 0, stream>>>(
      proj1, ei, logits, denom, out1, E, N);
  gat_elu_inplace<<<gblk(N * 192), B, 0, stream>>>(out1, N * 192);  // out1 -> h

  // ===== layer 2 ===========================================================
  {
    int tiles4 = mtiles * 4;               // Nout = 64
    gat_wmma_gemm_f32<<<(tiles4 + wavesPerBlk - 1) / wavesPerBlk, B, 0, stream>>>(
        out1, W2, nullptr, proj2, mtiles, 4, 192);
    int tiles1 = mtiles * 1;               // Nout = 16
    gat_wmma_gemm_f32<<<(tiles1 + wavesPerBlk - 1) / wavesPerBlk, B, 0, stream>>>(
        out1, res2, bias2, res2out, mtiles, 1, 192); // res2out = h@res2^T + b2
  }
  gat_fill_u32<<<gblk(N * 4), B, 0, stream>>>(maxenc, 0u, N * 4);
  gat_fill_u32<<<gblk(N * 4), B, 0, stream>>>((unsigned*)denom, 0u, N * 4);
  gat_fill_u32<<<gblk(N * 64), B, 0, stream>>>((unsigned*)out2, 0u, N * 64);

  gat_edge_logits_max<16><<<gblk(Ep * 4), B, 0, stream>>>(
      proj2, ei, eattr, att2, eb2, logits, maxenc, E, N);
  gat_edge_exp_sum<<<gblk(Ep * 4), B, 0, stream>>>(ei, logits, maxenc, denom, E, N);
  gat_edge_scatter<16><<<gblk(Ep * 64), B, 0, stream>>>(
      proj2, ei, logits, denom, out2, E, N);

  gat_finalize<<<gblk(N), B, 0, stream>>>(out2, res2out, zout, N);
}